// GNN_Bet_10127532884217
// MI455X (gfx1250) — compile-verified
//
#include <hip/hip_runtime.h>

typedef float v2f __attribute__((ext_vector_type(2)));
typedef float v4f __attribute__((ext_vector_type(4)));
typedef float v8f __attribute__((ext_vector_type(8)));

#define N_NODES 8192
#define FH 32                 // H
#define KT 64                 // K-tile staged in LDS (64 x 32 floats = 8KB)
#define WAVES_PER_BLOCK 8
#define ROWS_PER_WAVE 16
#define ROWS_PER_BLOCK (WAVES_PER_BLOCK * ROWS_PER_WAVE)   // 128
#define SPLITK 4
#define KSPAN (N_NODES / SPLITK)                            // 2048

// ---------------------------------------------------------------------------
// Big GEMM (split-K): Ypart[split][N,32] = adj[:, ks:ks+2048] @ Z[ks:ks+2048,:]
// One wave computes a 16x32 output tile (2 v8f accumulators) with
// V_WMMA_F32_16X16X4_F32. Z K-tiles staged in LDS interleaved by K-pairs so
// each B fragment is a single aligned ds_load_b64 (no repack movs).
// adj is streamed with non-temporal loads (256MB > L2, read 4x per branch).
// grid = (64 row-blocks, 4 K-splits) -> 256 blocks / 2048 waves in flight.
// ---------------------------------------------------------------------------
__global__ __launch_bounds__(256)
void adj_gemm_split(const float* __restrict__ adj,
                    const float* __restrict__ Z,
                    float* __restrict__ Ypart)
{
    // Zs[p*64 + 2*c + j] = Z[k0 + 2p + j][c]   (p = K-pair, c = col, j = 0/1)
    __shared__ __align__(16) float Zs[KT * FH];

    const int tid  = threadIdx.x;
    const int wave = tid >> 5;
    const int lane = tid & 31;
    const int m    = lane & 15;       // row (A) / col (B,C,D) within tile
    const int h    = lane >> 4;       // lane half selects K pair / row half
    const int rowBase = blockIdx.x * ROWS_PER_BLOCK + wave * ROWS_PER_WAVE;
    const int ks      = blockIdx.y * KSPAN;

    v8f acc0 = {};   // cols 0..15
    v8f acc1 = {};   // cols 16..31

    for (int k0 = ks; k0 < ks + KSPAN; k0 += KT) {
        __syncthreads();
        // Cooperative staging, interleaved-pair layout. 512 v4f writes.
#pragma unroll
        for (int it = 0; it < 2; ++it) {
            const int i  = tid + it * 256;       // 0..511
            const int p  = i >> 4;               // K-pair 0..31
            const int cp = (i & 15) * 2;         // column 0,2,..,30
            v2f a = *(const v2f*)(Z + (size_t)(k0 + 2 * p) * FH + cp);
            v2f b = *(const v2f*)(Z + (size_t)(k0 + 2 * p + 1) * FH + cp);
            v4f w; w.x = a.x; w.y = b.x; w.z = a.y; w.w = b.y;
            *(v4f*)(Zs + p * 64 + cp * 2) = w;
        }
        __syncthreads();

        const float* arow  = adj + (size_t)(rowBase + m) * N_NODES + k0 + 2 * h;
        const float* zbase = Zs + h * 64 + 2 * m;     // pair index kk/2 + h
#pragma unroll
        for (int kk = 0; kk < KT; kk += 4) {
            // A fragment (16x4 f32): lane holds adj[rowBase+m][k0+kk+2h .. +1]
            v2f a = __builtin_nontemporal_load((const v2f*)(arow + kk));
            // B fragments: {Z[kk+2h][n], Z[kk+2h+1][n]} as one b64 each
            const float* zp = zbase + (kk >> 1) * 64;
            v2f b0 = *(const v2f*)(zp);          // cols 0..15
            v2f b1 = *(const v2f*)(zp + 32);     // cols 16..31
            acc0 = __builtin_amdgcn_wmma_f32_16x16x4_f32(
                false, a, false, b0, (short)0, acc0, false, false);
            acc1 = __builtin_amdgcn_wmma_f32_16x16x4_f32(
                false, a, false, b1, (short)0, acc1, false, false);
        }
    }

    // Store raw partial sums; epilogue kernel reduces/relu/normalizes.
    float* yp = Ypart + (size_t)blockIdx.y * N_NODES * FH;
#pragma unroll
    for (int v = 0; v < 8; v++) {
        const int r = rowBase + v + 8 * h;
        yp[(size_t)r * FH + m]      = acc0[v];
        yp[(size_t)r * FH + 16 + m] = acc1[v];
    }
}

// ---------------------------------------------------------------------------
// Epilogue: Y = relu(sum_p Ypart[p]), optional per-row L2 normalize.
// One element/thread; each wave32 owns exactly one 32-wide row.
// ---------------------------------------------------------------------------
__global__ __launch_bounds__(256)
void reduce_relu_norm(const float* __restrict__ Ypart,
                      float* __restrict__ Y,
                      int doNorm)
{
    const int idx = blockIdx.x * 256 + threadIdx.x;   // element index
    float s = 0.0f;
#pragma unroll
    for (int p = 0; p < SPLITK; p++)
        s += Ypart[(size_t)p * N_NODES * FH + idx];
    s = fmaxf(s, 0.0f);
    if (doNorm) {
        float ss = s * s;
        ss += __shfl_xor(ss, 1, 32);
        ss += __shfl_xor(ss, 2, 32);
        ss += __shfl_xor(ss, 4, 32);
        ss += __shfl_xor(ss, 8, 32);
        ss += __shfl_xor(ss, 16, 32);
        s *= 1.0f / fmaxf(sqrtf(ss), 1e-12f);
    }
    Y[idx] = s;
}

// ---------------------------------------------------------------------------
// Small feature transform: Zout[N,32] = X[N,32] @ W[32,32]. One elem/thread.
// ---------------------------------------------------------------------------
__global__ __launch_bounds__(256)
void feat_transform(const float* __restrict__ X,
                    const float* __restrict__ W,
                    float* __restrict__ Zout)
{
    __shared__ __align__(16) float Ws[FH * FH];
    const int tid = threadIdx.x;
    ((v4f*)Ws)[tid] = ((const v4f*)W)[tid];   // 1024 floats = 256 float4
    __syncthreads();

    const int idx = blockIdx.x * 256 + tid;
    const int row = idx >> 5;
    const int col = idx & 31;
    const float* xr = X + (size_t)row * FH;
    float s = 0.0f;
#pragma unroll
    for (int k = 0; k < FH; k++)
        s += xr[k] * Ws[k * FH + col];
    Zout[idx] = s;
}

// ---------------------------------------------------------------------------
// Scoring MLP over all 8 feature maps; out = (sum s[0..3]) * (sum s[4..7]).
// ---------------------------------------------------------------------------
__global__ __launch_bounds__(256)
void mlp_score(const float* __restrict__ xs,   // [8][N][32] contiguous
               const float* __restrict__ l1w, const float* __restrict__ l1b,
               const float* __restrict__ l2w, const float* __restrict__ l2b,
               const float* __restrict__ l3w, const float* __restrict__ l3b,
               float* __restrict__ out)
{
    __shared__ float W1[FH * 2 * FH];          // 32x64
    __shared__ float W2[2 * FH * 2 * FH];      // 64x64
    __shared__ float W3[2 * FH];
    __shared__ float B1[2 * FH];
    __shared__ float B2[2 * FH];
    __shared__ float b3s;

    const int tid = threadIdx.x;
    for (int i = tid; i < FH * 2 * FH; i += 256)     W1[i] = l1w[i];
    for (int i = tid; i < 2 * FH * 2 * FH; i += 256) W2[i] = l2w[i];
    if (tid < 2 * FH) { W3[tid] = l3w[tid]; B1[tid] = l1b[tid]; B2[tid] = l2b[tid]; }
    if (tid == 0) b3s = l3b[0];
    __syncthreads();

    const int row = blockIdx.x * 256 + tid;
    float s1 = 0.0f, s2 = 0.0f;
    for (int k = 0; k < 8; k++) {
        const float* x = xs + ((size_t)k * N_NODES + row) * FH;
        float xl[FH];
#pragma unroll
        for (int i = 0; i < FH; i++) xl[i] = x[i];
        float h1[2 * FH];
#pragma unroll
        for (int j = 0; j < 2 * FH; j++) {
            float a = B1[j];
#pragma unroll
            for (int i = 0; i < FH; i++) a += xl[i] * W1[i * 2 * FH + j];
            h1[j] = fmaxf(a, 0.0f);
        }
        float s = b3s;
        for (int j = 0; j < 2 * FH; j++) {
            float a = B2[j];
#pragma unroll
            for (int i = 0; i < 2 * FH; i++) a += h1[i] * W2[i * 2 * FH + j];
            s += fmaxf(a, 0.0f) * W3[j];
        }
        if (k < 4) s1 += s; else s2 += s;
    }
    out[row] = s1 * s2;
}

// ---------------------------------------------------------------------------
extern "C" void kernel_launch(void* const* d_in, const int* in_sizes, int n_in,
                              void* d_out, int out_size, void* d_ws, size_t ws_size,
                              hipStream_t stream)
{
    const float* adj1 = (const float*)d_in[0];
    const float* adj2 = (const float*)d_in[1];
    const float* W0   = (const float*)d_in[2];    // [N,32] init-layer features
    const float* Wmid[3] = { (const float*)d_in[3],   // W2
                             (const float*)d_in[4],   // W3
                             (const float*)d_in[5] }; // W4
    const float* l1w = (const float*)d_in[6];
    const float* l1b = (const float*)d_in[7];
    const float* l2w = (const float*)d_in[8];
    const float* l2b = (const float*)d_in[9];
    const float* l3w = (const float*)d_in[10];
    const float* l3b = (const float*)d_in[11];
    float* out = (float*)d_out;

    // Workspace layout: xs[8][N][32] (8MB) | z[N][32] (1MB) | Ypart[4][N][32] (4MB)
    float* xs    = (float*)d_ws;
    float* z     = xs + (size_t)8 * N_NODES * FH;
    float* ypart = z + (size_t)N_NODES * FH;

    const dim3 gBig(N_NODES / ROWS_PER_BLOCK, SPLITK);   // (64, 4)
    const dim3 bBig(256);
    const dim3 gElem((N_NODES * FH) / 256);              // 1024 blocks
    const float* adjs[2] = { adj1, adj2 };

    for (int b = 0; b < 2; b++) {
        float* x0 = xs + (size_t)(4 * b) * N_NODES * FH;
        // layer 1: relu + normalize(adj @ W0)
        adj_gemm_split<<<gBig, bBig, 0, stream>>>(adjs[b], W0, ypart);
        reduce_relu_norm<<<gElem, 256, 0, stream>>>(ypart, x0, 1);
        for (int l = 0; l < 3; l++) {
            float* xin  = xs + (size_t)(4 * b + l) * N_NODES * FH;
            float* xout = xs + (size_t)(4 * b + l + 1) * N_NODES * FH;
            feat_transform<<<gElem, 256, 0, stream>>>(xin, Wmid[l], z);
            adj_gemm_split<<<gBig, bBig, 0, stream>>>(adjs[b], z, ypart);
            // layers 2,3 normalize; layer 4 relu only
            reduce_relu_norm<<<gElem, 256, 0, stream>>>(ypart, xout, l < 2 ? 1 : 0);
        }
    }
    mlp_score<<<N_NODES / 256, 256, 0, stream>>>(xs, l1w, l1b, l2w, l2b,
                                                 l3w, l3b, out);
}